// GraphAttentionLayer_40235253629558
// MI455X (gfx1250) — compile-verified
//
#include <hip/hip_runtime.h>

// GAT layer for MI455X (gfx1250, wave32).
//   B=4, N=2048, F_IN=256, H=4, FO=64
// Kernel A: h = x@W via V_WMMA_F32_16X16X4_F32, fused attn_src/attn_dst epilogue.
// Kernel B: flash-style masked softmax + P@V via V_WMMA_F32_16X16X4_F32.

typedef __attribute__((ext_vector_type(2))) float v2f;
typedef __attribute__((ext_vector_type(8))) float v8f;

#define B_   4
#define N_   2048
#define FIN_ 256
#define H_   4
#define FO_  64
#define HFO_ 256          // H*FO
#define M_   (B_ * N_)    // 8192 rows of x
#define NEG_BIG (-3.0e38f)

// ---------------------------------------------------------------------------
// Kernel A: one wave computes a 16x64 tile of h (cols = one head), 256 wmma.
//   grid = (M_/16, H_), block = 32
// A-frag (16x4 f32): lane<16 -> M=lane, K={0,1}; lane>=16 -> M=lane-16, K={2,3}
// B-frag (4x16 f32): lane<16 -> K={0,1} rows;   lane>=16 -> K={2,3} rows; N=lane&15
// C-frag (16x16):    vgpr r, lane<16 -> row r;  lane>=16 -> row r+8;      N=lane&15
// ---------------------------------------------------------------------------
__global__ __launch_bounds__(32) void gat_h_kernel(
    const float* __restrict__ x, const float* __restrict__ W,
    const float* __restrict__ a_src, const float* __restrict__ a_dst,
    float* __restrict__ h, float* __restrict__ asrc_out, float* __restrict__ adst_out)
{
    const int lane = threadIdx.x;
    const int mi   = lane & 15;
    const int khi  = (lane >> 4) << 1;      // 0 or 2 (K offset of this half)
    const int row0 = blockIdx.x * 16;       // global row in [0, 8192)
    const int head = blockIdx.y;
    const int f0   = head * FO_;            // 64-wide column chunk == one head

    v8f c[4] = {v8f{0}, v8f{0}, v8f{0}, v8f{0}};

    for (int k = 0; k < FIN_; k += 4) {
        const float* xp = x + (size_t)(row0 + mi) * FIN_ + (k + khi);
        v2f a; a.x = xp[0]; a.y = xp[1];
#pragma unroll
        for (int t = 0; t < 4; ++t) {
            const int n = f0 + t * 16 + mi;
            v2f bf;
            bf.x = W[(size_t)(k + khi)     * HFO_ + n];
            bf.y = W[(size_t)(k + khi + 1) * HFO_ + n];
            c[t] = __builtin_amdgcn_wmma_f32_16x16x4_f32(
                false, a, false, bf, (short)0, c[t], false, false);
        }
    }

    // attention-projection weights for this lane's 4 columns
    float ws_[4], wd_[4];
#pragma unroll
    for (int t = 0; t < 4; ++t) {
        ws_[t] = a_src[f0 + t * 16 + mi];
        wd_[t] = a_dst[f0 + t * 16 + mi];
    }

    const int half8 = (lane >> 4) << 3;     // row offset of this half
#pragma unroll
    for (int r = 0; r < 8; ++r) {
        const int row = row0 + r + half8;
        float rsrc = 0.f, rdst = 0.f;
#pragma unroll
        for (int t = 0; t < 4; ++t) {
            const float v = c[t][r];
            rsrc += v * ws_[t];
            rdst += v * wd_[t];
            h[(size_t)row * HFO_ + f0 + t * 16 + mi] = v;   // store h tile
        }
        // reduce over the 16 columns held by this half's lanes
#pragma unroll
        for (int m = 1; m < 16; m <<= 1) {
            rsrc += __shfl_xor(rsrc, m, 32);
            rdst += __shfl_xor(rdst, m, 32);
        }
        if (mi == 0) {      // lane0 -> rows 0..7, lane16 -> rows 8..15
            const int b = row / N_, n = row % N_;
            asrc_out[((size_t)b * N_ + n) * H_ + head] = rsrc;
            adst_out[((size_t)b * N_ + n) * H_ + head] = rdst;
        }
    }
}

// ---------------------------------------------------------------------------
// Kernel B: one wave per (b, head, 16-row query tile); online softmax over j,
// P@V accumulated with 16 wmma per 16-wide j tile (2048 wmma per wave).
//   grid = (N_/16, H_, B_), block = 32
// Lane owns query row i = i0 + (lane&15); its 8 logits are laid out exactly as
// the A-fragment of V_WMMA_F32_16X16X4_F32 (K offsets 4*kk + khi + {0,1}).
// ---------------------------------------------------------------------------
__global__ __launch_bounds__(32) void gat_attn_kernel(
    const float* __restrict__ adj, const float* __restrict__ h,
    const float* __restrict__ asrc, const float* __restrict__ adst,
    float* __restrict__ out)
{
    const int lane = threadIdx.x;
    const int mi   = lane & 15;
    const int khi  = (lane >> 4) << 1;      // 0 or 2
    const int half8 = (lane >> 4) << 3;
    const int i0   = blockIdx.x * 16;
    const int head = blockIdx.y;
    const int b    = blockIdx.z;

    const int   irow  = i0 + mi;
    const float src_i = asrc[((size_t)b * N_ + irow) * H_ + head];
    const float* __restrict__ adjR = adj + (size_t)irow * N_;
    const float* __restrict__ dstP = adst + (size_t)b * N_ * H_ + head;   // [j*H_]
    const float* __restrict__ hB   = h + (size_t)b * N_ * HFO_ + head * FO_; // V: hB[j*HFO_+f]

    float mrow = NEG_BIG;                   // running row max
    float lrow = 0.f;                       // running row sum
    v8f acc[4] = {v8f{0}, v8f{0}, v8f{0}, v8f{0}};

    for (int j0 = 0; j0 < N_; j0 += 16) {
        // ---- logits for this lane's 8 (i,j) entries, A-frag order ----
        float p[8];
        float tmax = NEG_BIG;
#pragma unroll
        for (int kk = 0; kk < 4; ++kk) {
            const int j = j0 + 4 * kk + khi;
            float s0 = src_i + dstP[(size_t)j * H_];
            float s1 = src_i + dstP[(size_t)(j + 1) * H_];
            s0 = (s0 > 0.f) ? s0 : 0.2f * s0;       // leaky_relu(0.2)
            s1 = (s1 > 0.f) ? s1 : 0.2f * s1;
            s0 = (adjR[j]     != 0.f) ? s0 : NEG_BIG;
            s1 = (adjR[j + 1] != 0.f) ? s1 : NEG_BIG;
            p[2 * kk]     = s0;
            p[2 * kk + 1] = s1;
            tmax = fmaxf(tmax, fmaxf(s0, s1));
        }
        tmax = fmaxf(tmax, __shfl_xor(tmax, 16, 32));   // both halves share row mi
        const float mnew  = fmaxf(mrow, tmax);
        const float scale = __expf(mrow - mnew);        // 1 if both NEG_BIG, 0 if jump

        float psum = 0.f;
#pragma unroll
        for (int q = 0; q < 8; ++q) {
            p[q] = __expf(p[q] - mnew);
            psum += p[q];
        }
        psum += __shfl_xor(psum, 16, 32);
        lrow = lrow * scale + psum;
        mrow = mnew;

        // ---- rescale accumulator rows (broadcast per-row scale via bpermute) ----
#pragma unroll
        for (int r = 0; r < 8; ++r) {
            const float sc = __shfl(scale, r + half8, 32);
#pragma unroll
            for (int t = 0; t < 4; ++t) acc[t][r] *= sc;
        }

        // ---- P (16x16) @ V (16x64) via 4 K-steps x 4 col-chunks ----
#pragma unroll
        for (int kk = 0; kk < 4; ++kk) {
            v2f a; a.x = p[2 * kk]; a.y = p[2 * kk + 1];
            const float* vp = hB + (size_t)(j0 + 4 * kk + khi) * HFO_;
#pragma unroll
            for (int t = 0; t < 4; ++t) {
                v2f bf;
                bf.x = vp[t * 16 + mi];
                bf.y = vp[HFO_ + t * 16 + mi];
                acc[t] = __builtin_amdgcn_wmma_f32_16x16x4_f32(
                    false, a, false, bf, (short)0, acc[t], false, false);
            }
        }
    }

    // ---- epilogue: divide each row by its softmax denominator, store ----
#pragma unroll
    for (int r = 0; r < 8; ++r) {
        const float li  = __shfl(lrow, r + half8, 32);
        const float inv = 1.0f / li;                    // self-loops => li > 0
        const int   row = i0 + r + half8;
#pragma unroll
        for (int t = 0; t < 4; ++t) {
            out[((size_t)b * N_ + row) * HFO_ + head * FO_ + t * 16 + mi] =
                acc[t][r] * inv;
        }
    }
}

// ---------------------------------------------------------------------------
extern "C" void kernel_launch(void* const* d_in, const int* in_sizes, int n_in,
                              void* d_out, int out_size, void* d_ws, size_t ws_size,
                              hipStream_t stream)
{
    const float* x     = (const float*)d_in[0];  // (B,N,F_IN)
    const float* adj   = (const float*)d_in[1];  // (N,N)
    const float* W     = (const float*)d_in[2];  // (F_IN, H*FO)
    const float* a_src = (const float*)d_in[3];  // (H,FO)
    const float* a_dst = (const float*)d_in[4];  // (H,FO)
    float*       out   = (float*)d_out;          // (B,N,H*FO)

    // workspace layout: h (8 MB) | attn_src (128 KB) | attn_dst (128 KB)
    float* h    = (float*)d_ws;
    float* asrc = h + (size_t)M_ * HFO_;
    float* adst = asrc + (size_t)M_ * H_;

    gat_h_kernel<<<dim3(M_ / 16, H_), 32, 0, stream>>>(
        x, W, a_src, a_dst, h, asrc, adst);
    gat_attn_kernel<<<dim3(N_ / 16, H_, B_), 32, 0, stream>>>(
        adj, h, asrc, adst, out);
}